// MoCo_23450521436401
// MI455X (gfx1250) — compile-verified
//
#include <hip/hip_runtime.h>
#include <hip/hip_bf16.h>
#include <cstddef>

typedef __attribute__((ext_vector_type(2))) float v2f;
typedef __attribute__((ext_vector_type(4))) float v4f;
typedef __attribute__((ext_vector_type(8))) float v8f;
typedef __attribute__((ext_vector_type(4))) int   v4i;

#define MMT_C      0.999f
#define INV_TEMP   (1.0f / 0.07f)
#define NEG_INF_C  (-9.0e15f)

// ---------------- momentum EMA: out = wk*m + wq*(1-m) ----------------
__global__ void k_momentum(const float* __restrict__ wq, const float* __restrict__ wk,
                           float* __restrict__ out, int n) {
  int i = blockIdx.x * 256 + threadIdx.x;
  if (i < n) out[i] = wk[i] * MMT_C + wq[i] * (1.0f - MMT_C);
}

__global__ void k_zero(float* __restrict__ p, int n) {
  int i = blockIdx.x * 256 + threadIdx.x;
  if (i < n) p[i] = 0.0f;
}

// ---------------- WMMA f32 GEMM: C = act(A[M,K] @ B[K,N] + bias) * scale ----------------
// One 16x16 output tile per wave; K stepped by 4 via V_WMMA_F32_16X16X4_F32.
__global__ void k_wmma_gemm(const float* __restrict__ A, const float* __restrict__ B,
                            const float* __restrict__ bias, float* __restrict__ C,
                            int M, int N, int K, int relu, float scale) {
  int wave = threadIdx.x >> 5;
  int lane = threadIdx.x & 31;
  int tilesN = N >> 4;
  int tilesTotal = (M >> 4) * tilesN;
  int tile = blockIdx.x * 8 + wave;
  if (tile >= tilesTotal) return;           // wave-uniform exit; EXEC stays all-ones
  int m0 = (tile / tilesN) << 4;
  int n0 = (tile % tilesN) << 4;
  int l15 = lane & 15, hi = lane >> 4;

  const float* arow = A + (size_t)(m0 + l15) * K + 2 * hi;
  const float* bptr = B + (size_t)(2 * hi) * N + n0 + l15;

  v8f c = {};
  for (int k0 = 0; k0 < K; k0 += 4) {
    v2f a, b;
    a.x = arow[k0];
    a.y = arow[k0 + 1];
    b.x = bptr[(size_t)k0 * N];
    b.y = bptr[(size_t)k0 * N + N];
    c = __builtin_amdgcn_wmma_f32_16x16x4_f32(false, a, false, b, (short)0, c, false, false);
  }

  int col = n0 + l15;
  float bv = bias ? bias[col] : 0.0f;
#pragma unroll
  for (int r = 0; r < 8; ++r) {
    int row = m0 + r + 8 * hi;
    float v = c[r] + bv;
    if (relu) v = fmaxf(v, 0.0f);
    C[(size_t)row * N + col] = v * scale;
  }
}

// ---------------- row L2 normalize, width 128 ----------------
__global__ void k_l2norm128(float* __restrict__ P) {
  __shared__ float red[4];
  int row = blockIdx.x, t = threadIdx.x;
  float v = P[(size_t)row * 128 + t];
  float s = v * v;
  for (int off = 16; off; off >>= 1) s += __shfl_down(s, off, 32);
  if ((t & 31) == 0) red[t >> 5] = s;
  __syncthreads();
  float inv = rsqrtf(red[0] + red[1] + red[2] + red[3]);
  P[(size_t)row * 128 + t] = v * inv;
}

// ---------------- enqueue plan: sequential ranks, columns, per-qid groups ----------------
__global__ void k_plan(const int* __restrict__ q_ids, const int* __restrict__ queue_ptr,
                       int* __restrict__ cols, int* __restrict__ counts,
                       int* __restrict__ gstart, int* __restrict__ rowlist) {
  __shared__ int sq[1024];
  __shared__ int scnt[256];
  __shared__ int sgs[256];
  int n = threadIdx.x;                       // 1024 threads, one block
  sq[n] = q_ids[n];
  if (n < 256) scnt[n] = 0;
  __syncthreads();
  int qid = sq[n];
  int r = 0;
  for (int m = 0; m < n; ++m) r += (sq[m] == qid) ? 1 : 0;   // sequential rank
  cols[n] = (queue_ptr[qid] + r) & 2047;
  atomicAdd(&scnt[qid], 1);
  __syncthreads();
  if (n == 0) {
    int acc = 0;
    for (int i = 0; i < 256; ++i) { sgs[i] = acc; acc += scnt[i]; }
  }
  __syncthreads();
  rowlist[sgs[qid] + r] = n;
  if (n < 256) { counts[n] = scnt[n]; gstart[n] = sgs[n]; }
}

// ---------------- gather rows + l_pos + labels ----------------
__global__ void k_gather(const float* __restrict__ qfull, const float* __restrict__ kfull,
                         const int* __restrict__ idxq, const int* __restrict__ idxk,
                         const int* __restrict__ q_ids,
                         float* __restrict__ Qg, float* __restrict__ Kg,
                         float* __restrict__ logits_local,
                         float* __restrict__ labels_local, float* __restrict__ labels_global) {
  __shared__ float red[4];
  int n = blockIdx.x, t = threadIdx.x;       // 128 threads
  float qv = qfull[(size_t)idxq[n] * 128 + t];
  float kv = kfull[(size_t)idxk[n] * 128 + t];
  Qg[(size_t)n * 128 + t] = qv;
  Kg[(size_t)n * 128 + t] = kv;
  float s = qv * kv;
  for (int off = 16; off; off >>= 1) s += __shfl_down(s, off, 32);
  if ((t & 31) == 0) red[t >> 5] = s;
  __syncthreads();
  if (t == 0) {
    logits_local[(size_t)n * 2049] = (red[0] + red[1] + red[2] + red[3]) * INV_TEMP;
    labels_local[n] = 0.0f;
    labels_global[n] = (float)q_ids[n];
  }
}

// ---------------- fused queue stream: copy -> new_queue, column-sum -> meanT,
//                  grouped l_neg_local dots, all in ONE pass over the 268MB queue ----------------
#define SRM 8
__global__ void k_stream(const float* __restrict__ queue, const int* __restrict__ ids,
                         const float* __restrict__ Qg, const int* __restrict__ elem_ids,
                         const int* __restrict__ counts, const int* __restrict__ gstart,
                         const int* __restrict__ rowlist,
                         float* __restrict__ new_queue, float* __restrict__ meanT,
                         float* __restrict__ logits_local) {
  __shared__ float qs[SRM][128];
  __shared__ float wpart[128][8];
  __shared__ int rows[SRM];
  int qid  = blockIdx.x >> 1;                       // 256 qids x 2 halves
  int col  = ((blockIdx.x & 1) << 10) + (threadIdx.x << 2);   // 4 cols per thread
  int cidx = col >> 2;                              // float4 index in a 2048-col row
  int lane = threadIdx.x & 31, wave = threadIdx.x >> 5;
  int cnt = counts[qid], gs = gstart[qid];
  const v4f* qsrc = (const v4f*)(queue     + (size_t)qid * 128 * 2048);
  v4f*       qdst = (v4f*)(new_queue       + (size_t)qid * 128 * 2048);
  v4i myids = ((const v4i*)(ids + (size_t)qid * 2048))[cidx];

  int passes = (cnt + SRM - 1) / SRM;
  if (passes < 1) passes = 1;                       // cnt==0 still copies + means
  for (int p = 0; p < passes; ++p) {
    int r0 = p * SRM;
    int nr = cnt - r0; if (nr > SRM) nr = SRM; if (nr < 0) nr = 0;
    __syncthreads();
    for (int i = threadIdx.x; i < SRM * 128; i += 256) {
      int ri = i >> 7, c = i & 127;
      if (ri < nr) {
        int n = rowlist[gs + r0 + ri];
        qs[ri][c] = Qg[(size_t)n * 128 + c];
        if (c == 0) rows[ri] = n;
      } else qs[ri][c] = 0.0f;
    }
    __syncthreads();

    v4f acc[SRM];
    v4f zero = {};
#pragma unroll
    for (int i = 0; i < SRM; ++i) acc[i] = zero;

    if (p == 0) {
      // streaming pass: NT load, NT store (write-once data; keep L2 for reused tiles)
      for (int c = 0; c < 128; ++c) {
        __builtin_prefetch(&qsrc[(size_t)(c + 4) * 512 + cidx], 0, 1);
        v4f v = __builtin_nontemporal_load(&qsrc[(size_t)c * 512 + cidx]);
        __builtin_nontemporal_store(v, &qdst[(size_t)c * 512 + cidx]);
#pragma unroll
        for (int i = 0; i < SRM; ++i) acc[i] += qs[i][c] * v;
        float wsum = v.x + v.y + v.z + v.w;
        for (int off = 16; off; off >>= 1) wsum += __shfl_down(wsum, off, 32);
        if (lane == 0) wpart[c][wave] = wsum;       // fixed slot: deterministic
      }
      __syncthreads();
      if (threadIdx.x < 128) {
        float s = 0.0f;
#pragma unroll
        for (int w = 0; w < 8; ++w) s += wpart[threadIdx.x][w];   // fixed order
        // exactly 2 blocks add per (c,qid): commutative -> bitwise deterministic
        atomicAdd(&meanT[(size_t)threadIdx.x * 256 + qid], s);
      }
    } else {
      // rare overflow pass (cnt > 8): read-only
      for (int c = 0; c < 128; ++c) {
        v4f v = qsrc[(size_t)c * 512 + cidx];
#pragma unroll
        for (int i = 0; i < SRM; ++i) acc[i] += qs[i][c] * v;
      }
    }

    for (int i = 0; i < nr; ++i) {
      int n = rows[i];
      int eid = elem_ids[n];
      float* dst = logits_local + (size_t)n * 2049 + 1 + col;
      dst[0] = ((myids.x == eid) ? NEG_INF_C : acc[i].x) * INV_TEMP;
      dst[1] = ((myids.y == eid) ? NEG_INF_C : acc[i].y) * INV_TEMP;
      dst[2] = ((myids.z == eid) ? NEG_INF_C : acc[i].z) * INV_TEMP;
      dst[3] = ((myids.w == eid) ? NEG_INF_C : acc[i].w) * INV_TEMP;
    }
  }
}

// ---------------- queue update outputs ----------------
__global__ void k_idscopy(const int* __restrict__ ids, float* __restrict__ out) {
  int i = blockIdx.x * 256 + threadIdx.x;
  if (i < 256 * 2048) out[i] = (float)ids[i];
}

__global__ void k_scatter(const float* __restrict__ Kg, const int* __restrict__ q_ids,
                          const int* __restrict__ cols, const int* __restrict__ elem_ids,
                          float* __restrict__ new_queue, float* __restrict__ new_ids) {
  int n = blockIdx.x, t = threadIdx.x;       // 128 threads per enqueue row
  int qid = q_ids[n], col = cols[n];
  new_queue[((size_t)qid * 128 + t) * 2048 + col] = Kg[(size_t)n * 128 + t];
  if (t == 0) new_ids[(size_t)qid * 2048 + col] = (float)elem_ids[n];
}

__global__ void k_ptr(const int* __restrict__ queue_ptr, const int* __restrict__ counts,
                      float* __restrict__ new_ptr) {
  int i = threadIdx.x;
  if (i < 256) new_ptr[i] = (float)((queue_ptr[i] + counts[i]) & 2047);
}

extern "C" void kernel_launch(void* const* d_in, const int* in_sizes, int n_in,
                              void* d_out, int out_size, void* d_ws, size_t ws_size,
                              hipStream_t stream) {
  const float* inputs_q  = (const float*)d_in[0];
  const float* inputs_k  = (const float*)d_in[1];
  const int*   idxq      = (const int*)d_in[2];
  const int*   idxk      = (const int*)d_in[3];
  const int*   q_ids     = (const int*)d_in[4];
  const int*   elem_ids  = (const int*)d_in[5];
  const float* wq1 = (const float*)d_in[6];
  const float* bq1 = (const float*)d_in[7];
  const float* wq2 = (const float*)d_in[8];
  const float* bq2 = (const float*)d_in[9];
  const float* wk1 = (const float*)d_in[10];
  const float* bk1 = (const float*)d_in[11];
  const float* wk2 = (const float*)d_in[12];
  const float* bk2 = (const float*)d_in[13];
  const float* queue     = (const float*)d_in[14];
  const int*   ids       = (const int*)d_in[15];
  const int*   queue_ptr = (const int*)d_in[16];

  float* out           = (float*)d_out;
  float* logits_local  = out;                 // 1024*2049
  float* labels_local  = out + 2098176;       // 1024
  float* logits_global = out + 2099200;       // 1024*256
  float* labels_global = out + 2361344;       // 1024
  float* new_queue     = out + 2362368;       // 256*128*2048
  float* new_ids       = out + 69471232;      // 256*2048
  float* new_ptr       = out + 69995520;      // 256

  float* ws    = (float*)d_ws;
  float* wk1m  = ws + 0;        // 65536
  float* bk1m  = ws + 65536;    // 256
  float* wk2m  = ws + 65792;    // 32768
  float* bk2m  = ws + 98560;    // 128
  float* Hq    = ws + 98688;    // 1048576
  float* Pq    = ws + 1147264;  // 524288 (normalized in place -> q_full)
  float* Hk    = ws + 1671552;  // 1048576
  float* Pk    = ws + 2720128;  // 524288 (-> k_full)
  float* Qg    = ws + 3244416;  // 131072
  float* Kg    = ws + 3375488;  // 131072
  float* meanT = ws + 3506560;  // 32768 (column-sums; /2048 folded into GEMM scale)
  int*   iws     = (int*)(ws + 3539328);
  int*   colsArr = iws;         // 1024
  int*   counts  = iws + 1024;  // 256
  int*   gstart  = iws + 1280;  // 256
  int*   rowlist = iws + 1536;  // 1024

  // 1) momentum update of key projector
  k_momentum<<<256, 256, 0, stream>>>(wq1, wk1, wk1m, 65536);
  k_momentum<<<1, 256, 0, stream>>>(bq1, bk1, bk1m, 256);
  k_momentum<<<128, 256, 0, stream>>>(wq2, wk2, wk2m, 32768);
  k_momentum<<<1, 256, 0, stream>>>(bq2, bk2, bk2m, 128);

  // 2) projections (WMMA f32), then row L2 norms
  k_wmma_gemm<<<512, 256, 0, stream>>>(inputs_q, wq1, bq1, Hq, 4096, 256, 256, 1, 1.0f);
  k_wmma_gemm<<<256, 256, 0, stream>>>(Hq, wq2, bq2, Pq, 4096, 128, 256, 0, 1.0f);
  k_l2norm128<<<4096, 128, 0, stream>>>(Pq);
  k_wmma_gemm<<<512, 256, 0, stream>>>(inputs_k, wk1m, bk1m, Hk, 4096, 256, 256, 1, 1.0f);
  k_wmma_gemm<<<256, 256, 0, stream>>>(Hk, wk2m, bk2m, Pk, 4096, 128, 256, 0, 1.0f);
  k_l2norm128<<<4096, 128, 0, stream>>>(Pk);

  // 3) enqueue plan, row gathers, l_pos, labels
  k_plan<<<1, 1024, 0, stream>>>(q_ids, queue_ptr, colsArr, counts, gstart, rowlist);
  k_gather<<<1024, 128, 0, stream>>>(Pq, Pk, idxq, idxk, q_ids, Qg, Kg,
                                     logits_local, labels_local, labels_global);

  // 4) single fused pass over the queue: copy + column-sums + local negatives
  k_zero<<<128, 256, 0, stream>>>(meanT, 32768);
  k_stream<<<512, 256, 0, stream>>>(queue, ids, Qg, elem_ids, counts, gstart, rowlist,
                                    new_queue, meanT, logits_local);

  // 5) logits_global = Qg @ meanTᵀ-sums * (1/(2048*TEMP))  (WMMA)
  k_wmma_gemm<<<128, 256, 0, stream>>>(Qg, meanT, nullptr, logits_global,
                                       1024, 256, 128, 0, INV_TEMP * (1.0f / 2048.0f));

  // 6) ids convert-copy, sparse column scatter of k, ptr update
  k_idscopy<<<2048, 256, 0, stream>>>(ids, new_ids);
  k_scatter<<<1024, 128, 0, stream>>>(Kg, q_ids, colsArr, elem_ids, new_queue, new_ids);
  k_ptr<<<1, 256, 0, stream>>>(queue_ptr, counts, new_ptr);
}